// NoisytopkRouter_609885356202
// MI455X (gfx1250) — compile-verified
//
#include <hip/hip_runtime.h>
#include <math.h>

// Noisy top-k router for B=8, S=4096, D=2048, E=64, K=8 on gfx1250 (MI455X).
// x (256 MB) streamed HBM -> LDS via GLOBAL_LOAD_ASYNC_TO_LDS_B128 with double
// buffering (DMA of chunk n+1 overlaps WMMA on chunk n, ASYNCcnt-tracked).
// W amortized over 64 rows/block (~512 MB L2 traffic). GEMM on
// V_WMMA_F32_16X16X4_F32, 4 independent accumulator chains share each B frag.

#define D_DIM 2048
#define E_DIM 64
#define K_TOP 8
#define ROWS_PER_BLOCK 64
#define KCHUNK 128
#define NCHUNK (D_DIM / KCHUNK)      // 16
#define XS_STRIDE (KCHUNK + 4)       // row stride % 64 banks == 4 -> conflict-free A reads

typedef __attribute__((ext_vector_type(2))) float v2f;
typedef __attribute__((ext_vector_type(8))) float v8f;

#define WMMA_F32(A, B, C) \
    __builtin_amdgcn_wmma_f32_16x16x4_f32(false, (A), false, (B), (short)0, (C), false, false)

__global__ __launch_bounds__(256) void noisy_topk_router_kernel(
    const float* __restrict__ x, const float* __restrict__ noise,
    const float* __restrict__ Wl, const float* __restrict__ bl,
    const float* __restrict__ Wn, const float* __restrict__ bn,
    float* __restrict__ out_probs, int* __restrict__ out_idx)
{
    __shared__ float xs[2][ROWS_PER_BLOCK * XS_STRIDE];       // 2 x 33 KB x-tile chunks
    __shared__ float lg[2][ROWS_PER_BLOCK][E_DIM];            // 32 KB logits/noise_logits

    const int tid  = threadIdx.x;
    const int lane = tid & 31;
    const int wv   = tid >> 5;              // wave 0..7
    const int mat  = wv >> 2;               // 0 = W_linear, 1 = W_noise
    const int eBase = (wv & 3) << 4;        // expert tile: 0,16,32,48
    const size_t row0 = (size_t)blockIdx.x * ROWS_PER_BLOCK;

    const int mrow = lane & 15;             // A: row M within tile / B: column N (expert)
    const int kOff = (lane >> 4) << 1;      // lanes 0-15 -> K+0..1, lanes 16-31 -> K+2..3

    const float* __restrict__ Wm   = mat ? Wn : Wl;
    const float* __restrict__ wrow = Wm + (size_t)(eBase + mrow) * D_DIM;

    // Async DMA one 64 x KCHUNK x-tile chunk into LDS buffer `buf` (no VGPR staging).
    auto issue_chunk = [&](int kc, int buf) {
        // 64 rows * 32 float4 per row = 2048 b128 transfers, 8 per thread.
        #pragma unroll
        for (int j = 0; j < 8; ++j) {
            int i = j * 256 + tid;
            int r = i >> 5;                 // i / 32 : row 0..63
            int c = (i & 31) << 2;          // float column (16B granules), coalesced
            unsigned ldsoff = (unsigned)(size_t)&xs[buf][r * XS_STRIDE + c];
            const float* g = x + (row0 + r) * D_DIM + kc + c;
            asm volatile("global_load_async_to_lds_b128 %0, %1, off"
                         :: "v"(ldsoff), "v"(g) : "memory");
        }
    };

    v8f acc0 = {};   // rows  0..15
    v8f acc1 = {};   // rows 16..31
    v8f acc2 = {};   // rows 32..47
    v8f acc3 = {};   // rows 48..63

    issue_chunk(0, 0);
    asm volatile("s_wait_asynccnt 0x0" ::: "memory");
    __syncthreads();                         // chunk 0 resident for all waves

    for (int n = 0; n < NCHUNK; ++n) {
        const int kc  = n * KCHUNK;
        const int cur = n & 1;

        // Overlap: DMA chunk n+1 into the other buffer while we do WMMA on chunk n.
        if (n + 1 < NCHUNK) issue_chunk(kc + KCHUNK, (n + 1) & 1);

        // ---- WMMA f32 16x16x4: one B fragment feeds 4 row tiles ----
        const float* wr  = wrow + kc + kOff;                             // W[e,k..k+1] (L2-hot)
        const float* xr0 = &xs[cur][( 0 + mrow) * XS_STRIDE + kOff];
        const float* xr1 = &xs[cur][(16 + mrow) * XS_STRIDE + kOff];
        const float* xr2 = &xs[cur][(32 + mrow) * XS_STRIDE + kOff];
        const float* xr3 = &xs[cur][(48 + mrow) * XS_STRIDE + kOff];
        #pragma unroll 4
        for (int kk = 0; kk < KCHUNK; kk += 4) {
            v2f b  = *(const v2f*)(wr  + kk);
            v2f a0 = *(const v2f*)(xr0 + kk);
            v2f a1 = *(const v2f*)(xr1 + kk);
            v2f a2 = *(const v2f*)(xr2 + kk);
            v2f a3 = *(const v2f*)(xr3 + kk);
            acc0 = WMMA_F32(a0, b, acc0);
            acc1 = WMMA_F32(a1, b, acc1);
            acc2 = WMMA_F32(a2, b, acc2);
            acc3 = WMMA_F32(a3, b, acc3);
        }

        if (n + 1 < NCHUNK) {
            asm volatile("s_wait_asynccnt 0x0" ::: "memory");   // my chunk n+1 DMA landed
            __syncthreads();                                    // everyone's landed / buf n consumed
        }
    }

    // ---- bias add, C tiles -> LDS ----
    // C layout: VGPR rr holds M=rr (lanes 0-15) / M=rr+8 (lanes 16-31), N = lane&15.
    const float biasv = (mat ? bn : bl)[eBase + mrow];
    #pragma unroll
    for (int rr = 0; rr < 8; ++rr) {
        int row = (lane < 16) ? rr : (rr + 8);
        lg[mat][ 0 + row][eBase + mrow] = acc0[rr] + biasv;
        lg[mat][16 + row][eBase + mrow] = acc1[rr] + biasv;
        lg[mat][32 + row][eBase + mrow] = acc2[rr] + biasv;
        lg[mat][48 + row][eBase + mrow] = acc3[rr] + biasv;
    }
    __syncthreads();

    // ---- per-row: noisy gate, top-8, softmax, scatter (one thread per row) ----
    if (tid < ROWS_PER_BLOCK) {
        const int r = tid;
        const size_t gRow = row0 + r;
        const float* nz = noise + gRow * E_DIM;

        for (int e = 0; e < E_DIM; ++e) {
            float zl = lg[0][r][e];
            float zn = lg[1][r][e];
            // numerically stable softplus
            float sp = fmaxf(zn, 0.0f) + log1pf(expf(-fabsf(zn)));
            lg[0][r][e] = zl + nz[e] * sp;    // noisy logits (reuse LDS)
        }

        float vals[K_TOP];
        int   idx[K_TOP];
        #pragma unroll
        for (int kk = 0; kk < K_TOP; ++kk) {
            float best = -INFINITY;
            int bi = 0;
            for (int e = 0; e < E_DIM; ++e) {
                float v = lg[0][r][e];
                if (v > best) { best = v; bi = e; }   // strict > == stable lowest-index
            }
            vals[kk] = best;
            idx[kk]  = bi;
            lg[0][r][bi] = -INFINITY;
        }

        // softmax over the K survivors (equals softmax of the -inf-filled sparse row)
        float mx = vals[0];
        float ex[K_TOP];
        float sum = 0.0f;
        #pragma unroll
        for (int kk = 0; kk < K_TOP; ++kk) { ex[kk] = expf(vals[kk] - mx); sum += ex[kk]; }
        float inv = 1.0f / sum;

        float* orow = out_probs + gRow * E_DIM;
        for (int e = 0; e < E_DIM; ++e) orow[e] = 0.0f;
        #pragma unroll
        for (int kk = 0; kk < K_TOP; ++kk) orow[idx[kk]] = ex[kk] * inv;

        int* irow = out_idx + gRow * K_TOP;
        #pragma unroll
        for (int kk = 0; kk < K_TOP; ++kk) irow[kk] = idx[kk];
    }
}

extern "C" void kernel_launch(void* const* d_in, const int* in_sizes, int n_in,
                              void* d_out, int out_size, void* d_ws, size_t ws_size,
                              hipStream_t stream)
{
    const float* x     = (const float*)d_in[0];
    const float* noise = (const float*)d_in[1];
    const float* Wl    = (const float*)d_in[2];
    const float* bl    = (const float*)d_in[3];
    const float* Wn    = (const float*)d_in[4];
    const float* bn    = (const float*)d_in[5];

    const size_t nrows = (size_t)in_sizes[1] / E_DIM;   // B*S = 32768 (from noise [B,S,E])

    float* out_probs = (float*)d_out;                         // [B,S,E] f32
    int*   out_idx   = (int*)(out_probs + nrows * E_DIM);     // [B,S,K] i32, concatenated

    dim3 grid((unsigned)(nrows / ROWS_PER_BLOCK));
    noisy_topk_router_kernel<<<grid, 256, 0, stream>>>(
        x, noise, Wl, bl, Wn, bn, out_probs, out_idx);
}